// IRT_24077586662107
// MI455X (gfx1250) — compile-verified
//
#include <hip/hip_runtime.h>
#include <math.h>

#define B_TOT 16384
#define NB    4            // batches per workgroup
#define ROWS  56           // NB * (6 + 8)
#define YSTR  260          // padded LDS row stride (floats) to avoid bank aliasing

typedef float v2f __attribute__((ext_vector_type(2)));
typedef float v8f __attribute__((ext_vector_type(8)));

__device__ __forceinline__ float redSum32(float v) {
#pragma unroll
    for (int m = 16; m >= 1; m >>= 1) v += __shfl_xor(v, m, 32);
    return v;
}
__device__ __forceinline__ float redSum8(float v) {
    v += __shfl_xor(v, 1, 32);
    v += __shfl_xor(v, 2, 32);
    v += __shfl_xor(v, 4, 32);
    return v;
}
__device__ __forceinline__ float redMax8(float v) {
    v = fmaxf(v, __shfl_xor(v, 1, 32));
    v = fmaxf(v, __shfl_xor(v, 2, 32));
    v = fmaxf(v, __shfl_xor(v, 4, 32));
    return v;
}

__global__ __launch_bounds__(128)
void immune_fused_kernel(const float* __restrict__ E,
                         const float* __restrict__ Kp,
                         const float* __restrict__ danger,
                         const float* __restrict__ w_prev,
                         const float* __restrict__ fitness,
                         const float* __restrict__ crisis,
                         const float* __restrict__ proto_conf,
                         const float* __restrict__ Lm,
                         const float* __restrict__ sigs,
                         float* __restrict__ out_w,
                         float* __restrict__ out_P) {
    __shared__ float Y[ROWS * YSTR];   // X @ L^T for this WG's 4 batches

    const int lane = threadIdx.x & 31;
    const int wave = threadIdx.x >> 5;
    const int b0   = blockIdx.x * NB;

    // ---------------- Phase 1: Y = X @ L^T via v_wmma_f32_16x16x4_f32 ----------------
    {
        // A fragment (16x4 f32): lanes 0-15 -> M=lane, K={k0,k0+1}; lanes 16-31 -> M=lane-16, K={k0+2,k0+3}
        int rowA = wave * 16 + (lane & 15);
        int rowC = (rowA < ROWS) ? rowA : 0;              // pad rows read row 0 (results masked at store)
        int bi = rowC / 14, j = rowC % 14;
        const float* aRow = (j < 6)
            ? (E  + (((size_t)(b0 + bi)) * 6 + j) * 256)
            : (Kp + (((size_t)(b0 + bi)) * 8 + (j - 6)) * 256);
        const int kOff = (lane >> 4) << 1;                // 0 or 2
        const int colL = lane & 15;

#pragma unroll 1
        for (int nt = 0; nt < 16; ++nt) {
            // B fragment (4x16 f32) of L^T: B[k,n] = L[n,k]; lane's column row of L is fixed
            const float* bRow = Lm + (size_t)(nt * 16 + colL) * 256;
            v8f acc0 = {0.f,0.f,0.f,0.f,0.f,0.f,0.f,0.f};
            v8f acc1 = {0.f,0.f,0.f,0.f,0.f,0.f,0.f,0.f};
#pragma unroll
            for (int k0 = 0; k0 < 256; k0 += 8) {
                v2f a0 = *(const v2f*)(aRow + k0 + kOff);
                v2f bb0 = *(const v2f*)(bRow + k0 + kOff);
                acc0 = __builtin_amdgcn_wmma_f32_16x16x4_f32(
                    false, a0, false, bb0, (short)0, acc0, false, false);
                v2f a1 = *(const v2f*)(aRow + k0 + 4 + kOff);
                v2f bb1 = *(const v2f*)(bRow + k0 + 4 + kOff);
                acc1 = __builtin_amdgcn_wmma_f32_16x16x4_f32(
                    false, a1, false, bb1, (short)0, acc1, false, false);
            }
            // C layout: VGPR r -> M=r (lanes 0-15) / M=r+8 (lanes 16-31), N = lane%16
            int col   = nt * 16 + colL;
            int rbase = wave * 16 + ((lane >> 4) << 3);
#pragma unroll
            for (int r = 0; r < 8; ++r) {
                int row = rbase + r;
                if (row < ROWS) Y[row * YSTR + col] = acc0[r] + acc1[r];
            }
        }
    }
    __syncthreads();

    // ---------------- Phase 2: one wave per batch ----------------
    const int b  = b0 + wave;
    const int nl = lane & 7;
    const float* Yb = &Y[(wave * 14) * YSTR];

    // EL rows in registers
    float ye[6][8];
#pragma unroll
    for (int m = 0; m < 6; ++m)
#pragma unroll
        for (int i = 0; i < 8; ++i)
            ye[m][i] = Yb[m * YSTR + lane + 32 * i];

    // ||EL||^2
    float sEL2[6];
#pragma unroll
    for (int m = 0; m < 6; ++m) {
        float p = 0.f;
#pragma unroll
        for (int i = 0; i < 8; ++i) p += ye[m][i] * ye[m][i];
        sEL2[m] = redSum32(p);
    }

    // ||KL||^2 and EL.KL cross terms, landed lane-per-column (nl)
    float crossL[6] = {0.f,0.f,0.f,0.f,0.f,0.f};
    float sKL2L = 0.f;
#pragma unroll
    for (int n = 0; n < 8; ++n) {
        float yk[8];
#pragma unroll
        for (int i = 0; i < 8; ++i) yk[i] = Yb[(6 + n) * YSTR + lane + 32 * i];
        float pk = 0.f;
        float pc[6] = {0.f,0.f,0.f,0.f,0.f,0.f};
#pragma unroll
        for (int i = 0; i < 8; ++i) {
            float kv = yk[i];
            pk += kv * kv;
#pragma unroll
            for (int m = 0; m < 6; ++m) pc[m] += ye[m][i] * kv;
        }
        pk = redSum32(pk);
        if (nl == n) sKL2L = pk;
#pragma unroll
        for (int m = 0; m < 6; ++m) {
            float cv = redSum32(pc[m]);
            if (nl == n) crossL[m] = cv;
        }
    }

    // self-signature norms
    float snrm[4];
#pragma unroll
    for (int s = 0; s < 4; ++s) {
        float p = 0.f;
#pragma unroll
        for (int i = 0; i < 8; ++i) {
            float sv = sigs[s * 256 + lane + 32 * i];
            p += sv * sv;
        }
        snrm[s] = sqrtf(redSum32(p));
    }

    // co-stimulation + tolerance penalty (from raw E)
    const float* dgr = danger + (size_t)b * 256;
    float co[6], tol[6];
#pragma unroll
    for (int m = 0; m < 6; ++m) {
        const float* er = E + ((size_t)b * 6 + m) * 256;
        float e8[8];
#pragma unroll
        for (int i = 0; i < 8; ++i) e8[i] = er[lane + 32 * i];
        float pd = 0.f, pn = 0.f;
#pragma unroll
        for (int i = 0; i < 8; ++i) {
            pd += e8[i] * dgr[lane + 32 * i];
            pn += e8[i] * e8[i];
        }
        co[m] = redSum32(pd);
        float en = sqrtf(redSum32(pn));
        float wm = -3.4e38f;
#pragma unroll
        for (int s = 0; s < 4; ++s) {
            float p = 0.f;
#pragma unroll
            for (int i = 0; i < 8; ++i) p += e8[i] * sigs[s * 256 + lane + 32 * i];
            float dv = redSum32(p);
            float cs = dv / (fmaxf(en, 1e-12f) * fmaxf(snrm[s], 1e-12f));
            wm = fmaxf(wm, cs);
        }
        tol[m] = fmaxf(wm - 0.1f, 0.0f);   // relu(KAPPA*worst - EPS_TOL), KAPPA=1
    }

    // proto self-similarity (from raw K), lane-per-column
    float pssL = 0.f;
#pragma unroll
    for (int n = 0; n < 8; ++n) {
        const float* kr = Kp + ((size_t)b * 8 + n) * 256;
        float k8[8];
#pragma unroll
        for (int i = 0; i < 8; ++i) k8[i] = kr[lane + 32 * i];
        float pn = 0.f;
#pragma unroll
        for (int i = 0; i < 8; ++i) pn += k8[i] * k8[i];
        float kn = sqrtf(redSum32(pn));
        float wm = -3.4e38f;
#pragma unroll
        for (int s = 0; s < 4; ++s) {
            float p = 0.f;
#pragma unroll
            for (int i = 0; i < 8; ++i) p += k8[i] * sigs[s * 256 + lane + 32 * i];
            float dv = redSum32(p);
            float cs = dv / (fmaxf(kn, 1e-12f) * fmaxf(snrm[s], 1e-12f));
            wm = fmaxf(wm, cs);
        }
        if (nl == n) pssL = wm;
    }

    // cost matrix, lane nl holds column nl for all 6 rows
    float pcf = proto_conf[(size_t)b * 8 + nl];
    float logKm[6];
#pragma unroll
    for (int m = 0; m < 6; ++m) {
        float mah = sEL2[m] + sKL2L - 2.0f * crossL[m];
        mah = fmaxf(mah, 1e-8f);
        float dist = sqrtf(mah);
        float Cc = dist - 0.5f * co[m] + 2.0f * tol[m] + 0.3f * pcf;
        logKm[m] = -Cc / (0.05f + 1e-8f);
    }

    // Sinkhorn with frozen-carry early stop (exact reference semantics)
    const float log_u = logf(1.0f / 6.0f + 1e-8f);
    const float log_v = logf(1.0f / 8.0f + 1e-8f);
    float la[6] = {0.f,0.f,0.f,0.f,0.f,0.f};
    float lbv = 0.0f;
    bool done = false;
    for (int it = 0; it < 10; ++it) {
        float na[6];
#pragma unroll
        for (int m = 0; m < 6; ++m) {
            float x = logKm[m] + lbv;               // uses current (possibly frozen) lb
            float mx = redMax8(x);
            float s = redSum8(expf(x - mx));
            na[m] = log_u - (mx + logf(s));
        }
        // nb: local logsumexp over m for this lane's column (uses fresh na)
        float x0 = logKm[0] + na[0];
        float mx2 = x0;
#pragma unroll
        for (int m = 1; m < 6; ++m) mx2 = fmaxf(mx2, logKm[m] + na[m]);
        float s2 = 0.f;
#pragma unroll
        for (int m = 0; m < 6; ++m) s2 += expf(logKm[m] + na[m] - mx2);
        float nb = log_v - (mx2 + logf(s2));

        float err = 0.f;
#pragma unroll
        for (int m = 0; m < 6; ++m) err = fmaxf(err, fabsf(na[m] - la[m]));
        if (!done) {
#pragma unroll
            for (int m = 0; m < 6; ++m) la[m] = na[m];
            lbv = nb;
        }
        done = done || ((it > 0) && (err < 1e-3f));
    }

    // P = clip(exp(la + logK + lb), 0, 1); p_mass = column sums
    float pmass = 0.f;
#pragma unroll
    for (int m = 0; m < 6; ++m) {
        float pv = expf(la[m] + logKm[m] + lbv);
        pv = fminf(pv, 1.0f);
        pmass += pv;
        if (lane < 8) out_P[(size_t)b * 48 + m * 8 + nl] = pv;
    }

    // replicator update (8 lanes)
    float wp = w_prev[(size_t)b * 8 + nl];
    float ft = fitness[(size_t)b * 8 + nl];
    float baseline = redSum8(wp * ft);
    float eta = 0.05f + 0.1f * crisis[b];
    float lt = logf(wp + 1e-8f) + eta * (ft - baseline) - 0.5f * pssL;
    float mx = redMax8(lt);
    float ex = expf(lt - mx);
    float ssum = redSum8(ex);
    float tld = ex / ssum;
    float wv = 0.7f * tld + 0.3f * pmass;
    float s2 = redSum8(wv) + 1e-8f;
    wv = wv / s2;
    wv = fminf(fmaxf(wv, 1e-6f), 1.0f);
    if (lane < 8) out_w[(size_t)b * 8 + nl] = wv;
}

extern "C" void kernel_launch(void* const* d_in, const int* in_sizes, int n_in,
                              void* d_out, int out_size, void* d_ws, size_t ws_size,
                              hipStream_t stream) {
    (void)in_sizes; (void)n_in; (void)out_size; (void)d_ws; (void)ws_size;
    const float* E          = (const float*)d_in[0];
    const float* Kp         = (const float*)d_in[1];
    const float* danger     = (const float*)d_in[2];
    const float* w_prev     = (const float*)d_in[3];
    const float* fitness    = (const float*)d_in[4];
    const float* crisis     = (const float*)d_in[5];
    const float* proto_conf = (const float*)d_in[6];
    const float* metric_L   = (const float*)d_in[7];
    const float* self_sigs  = (const float*)d_in[8];

    float* out_w = (float*)d_out;                      // [B, 8]
    float* out_P = out_w + (size_t)B_TOT * 8;          // [B, 6, 8]

    dim3 grid(B_TOT / NB);
    dim3 blk(128);
    immune_fused_kernel<<<grid, blk, 0, stream>>>(
        E, Kp, danger, w_prev, fitness, crisis, proto_conf,
        metric_L, self_sigs, out_w, out_P);
}